// RPN_21758304322176
// MI455X (gfx1250) — compile-verified
//
#include <hip/hip_runtime.h>
#include <math.h>

// ---------------------------------------------------------------------------
// RPN proposal generation for MI455X (gfx1250, wave32).
// Pipeline: per-level radix top-k -> gather+decode+clip -> greedy NMS -> emit.
// No matrix ops exist in this workload (IoU = per-pair max/min, not a GEMM),
// so no WMMA is used; instead the kernel uses the CDNA5-specific paths that
// fit: wave32 shuffle reductions, LDS, global_prefetch_b8, and the Tensor
// Data Mover (tensor_load_to_lds + s_wait_tensorcnt) for global->LDS staging.
// ---------------------------------------------------------------------------

#define A_TOTAL   159882
#define NIMG      16
#define KTOT      4507          // 1000+1000+1000+1000+507 selected per image
#define POSTN     1000
#define NMS_THR   0.7f
#define MIN_SIZE  1.0f
#define BBOX_CLIP 4.135166556742356f   // log(1000/16)

__constant__ int c_nPerLvl[5] = {120000, 30000, 7500, 1875, 507};
__constant__ int c_lvlOff[5]  = {0, 120000, 150000, 157500, 159375};
__constant__ int c_selOff[5]  = {0, 1000, 2000, 3000, 4000};
__constant__ int c_selCnt[5]  = {1000, 1000, 1000, 1000, 507};

typedef unsigned int u32x4 __attribute__((ext_vector_type(4)));
typedef unsigned int u32x8 __attribute__((ext_vector_type(8)));

// Monotonic float -> uint key (descending select on keys == descending floats)
__device__ __forceinline__ unsigned f2key(float f) {
    unsigned u = __float_as_uint(f);
    return (u & 0x80000000u) ? ~u : (u | 0x80000000u);
}

// ---------------------------------------------------------------------------
// Kernel 1: exact top-k per (image, level) via 4-pass 8-bit MSB radix select.
// One 1024-thread workgroup per (image, level). LDS histogram; streaming
// global reads with explicit (in-bounds) prefetch -> global_prefetch_b8.
// ---------------------------------------------------------------------------
__global__ __launch_bounds__(1024) void k_topk(const float* __restrict__ cls,
                                               int* __restrict__ sel) {
    const int img = blockIdx.x / 5;
    const int lvl = blockIdx.x % 5;
    const int n   = c_nPerLvl[lvl];
    const int off = c_lvlOff[lvl];
    const int k   = c_selCnt[lvl];
    const float* __restrict__ base = cls + (size_t)img * A_TOTAL + off;
    int* __restrict__ out = sel + img * KTOT + c_selOff[lvl];
    const int tid = threadIdx.x, bdim = blockDim.x;

    if (k >= n) {                       // level 4: everything survives
        for (int i = tid; i < n; i += bdim) out[i] = off + i;
        return;
    }

    __shared__ unsigned hist[256];
    __shared__ int s_bin, s_cum;
    __shared__ unsigned s_gt, s_eq;

    unsigned prefix = 0;
    int kRem = k;
    for (int shift = 24; shift >= 0; shift -= 8) {
        for (int b = tid; b < 256; b += bdim) hist[b] = 0;
        __syncthreads();
        const unsigned hm = (shift == 24) ? 0u : (0xFFFFFFFFu << (shift + 8));
        for (int i = tid; i < n; i += bdim) {
            int pf = i + 8 * bdim;
            if (pf < n) __builtin_prefetch(base + pf, 0, 0);  // global_prefetch_b8
            unsigned key = f2key(base[i]);
            if ((key & hm) == prefix)
                atomicAdd(&hist[(key >> shift) & 255u], 1u);
        }
        __syncthreads();
        if (tid == 0) {                  // 256-entry scan from the top
            int cum = 0, b = 255;
            for (; b > 0; --b) {
                int h = (int)hist[b];
                if (cum + h >= kRem) break;
                cum += h;
            }
            s_bin = b; s_cum = cum;
        }
        __syncthreads();
        prefix |= ((unsigned)s_bin) << shift;
        kRem  -= s_cum;
        __syncthreads();
    }

    // prefix == k-th largest key T; take all key>T plus kRem ties at T.
    if (tid == 0) { s_gt = 0; s_eq = 0; }
    __syncthreads();
    const unsigned T   = prefix;
    const unsigned nGT = (unsigned)(k - kRem);
    for (int i = tid; i < n; i += bdim) {
        int pf = i + 8 * bdim;
        if (pf < n) __builtin_prefetch(base + pf, 0, 0);
        unsigned key = f2key(base[i]);
        if (key > T) {
            unsigned p = atomicAdd(&s_gt, 1u);
            out[p] = off + i;
        } else if (key == T) {
            unsigned p = atomicAdd(&s_eq, 1u);
            if (p < (unsigned)kRem) out[nGT + p] = off + i;
        }
    }
}

// ---------------------------------------------------------------------------
// Kernel 2: gather + BoxCoder.decode + clip + validity for selected anchors
// only (16*4507 boxes instead of 16*159882 in the reference).
// ---------------------------------------------------------------------------
__global__ void k_init_max(unsigned* maxcU) {
    int t = threadIdx.x;
    if (t < NIMG) maxcU[t] = 0u;       // coords clipped >= 0, so 0-bits is min
}

__global__ void k_decode(const float* __restrict__ cls,
                         const float* __restrict__ reg,
                         const float* __restrict__ anchors,
                         const float* __restrict__ imgsz,
                         const int*   __restrict__ sel,
                         float* __restrict__ boxesG,
                         float* __restrict__ sigS,
                         float* __restrict__ smS,
                         unsigned* __restrict__ maxcU) {
    int t = blockIdx.x * blockDim.x + threadIdx.x;
    if (t >= NIMG * KTOT) return;
    const int img  = t / KTOT;
    const int aidx = sel[t];

    float raw = cls[(size_t)img * A_TOTAL + aidx];
    float sig = 1.0f / (1.0f + expf(-raw));

    const float4 r = ((const float4*)reg)[(size_t)img * A_TOTAL + aidx];
    const float4 a = ((const float4*)anchors)[aidx];
    float w  = a.z - a.x, h  = a.w - a.y;
    float cx = a.x + 0.5f * w, cy = a.y + 0.5f * h;
    float dw = fminf(r.z, BBOX_CLIP), dh = fminf(r.w, BBOX_CLIP);
    float pcx = r.x * w + cx, pcy = r.y * h + cy;
    float pw  = expf(dw) * w, ph  = expf(dh) * h;
    float x1 = pcx - 0.5f * pw, y1 = pcy - 0.5f * ph;
    float x2 = pcx + 0.5f * pw, y2 = pcy + 0.5f * ph;

    float ih = imgsz[img * 2 + 0], iw = imgsz[img * 2 + 1];
    x1 = fminf(fmaxf(x1, 0.0f), iw);  x2 = fminf(fmaxf(x2, 0.0f), iw);
    y1 = fminf(fmaxf(y1, 0.0f), ih);  y2 = fminf(fmaxf(y2, 0.0f), ih);

    bool valid = ((x2 - x1) >= MIN_SIZE) && ((y2 - y1) >= MIN_SIZE) && (sig >= 0.0f);

    float4 ob; ob.x = x1; ob.y = y1; ob.z = x2; ob.w = y2;
    ((float4*)boxesG)[t] = ob;
    sigS[t] = sig;
    smS[t]  = valid ? sig : -INFINITY;

    float mx = fmaxf(fmaxf(x1, y1), fmaxf(x2, y2));
    atomicMax(&maxcU[img], __float_as_uint(mx));   // nonneg floats: bit-monotonic
}

// ---------------------------------------------------------------------------
// Kernel 3: greedy batched-NMS, one workgroup (32 wave32 waves) per image.
// The mutable score vector is staged global->LDS by the Tensor Data Mover
// (tensor_load_to_lds, TENSORcnt-tracked); areas + level offsets also live in
// LDS (~54 KB of the 320 KB WGP pool). Block argmax uses wave32 shuffles.
// Areas/IoU are computed on the level-offset coordinates in the reference's
// exact op order so the iou>0.7 decisions match bit-for-bit.
// ---------------------------------------------------------------------------
__global__ __launch_bounds__(1024) void k_nms(const float* __restrict__ boxesG,
                                              const float* __restrict__ smS,
                                              const unsigned* __restrict__ maxcU,
                                              int* __restrict__ keep) {
    const int img = blockIdx.x;
    const int tid = threadIdx.x;
    const int bdim = blockDim.x;

    __shared__ float s_sc[KTOT];
    __shared__ float s_area[KTOT];
    __shared__ float s_off[KTOT];
    __shared__ float s_redV[32];
    __shared__ int   s_redI[32];
    __shared__ int   s_bestI;
    __shared__ float s_bestV;
    __shared__ float s_b[5];   // bx1,by1,bx2,by2,barea (offset coords)

    const float maxc = __uint_as_float(maxcU[img]);
    const float4* __restrict__ bb = (const float4*)boxesG + (size_t)img * KTOT;

    // ---- TDM: DMA smS[img*KTOT .. +KTOT) into s_sc as one 1-D tile ----
    if (tid < 32) {                                  // wave 0 issues the DMA
        unsigned long long ga =
            (unsigned long long)(uintptr_t)(smS + (size_t)img * KTOT);
        unsigned ldsa = (unsigned)(uintptr_t)(void*)s_sc;  // low 32b = LDS offset

        u32x4 g0;                                    // D# group 0 (ISA 8.3)
        g0[0] = 1u;                                  // count=1, user mode
        g0[1] = ldsa;                                // lds_addr
        g0[2] = (unsigned)(ga & 0xFFFFFFFFull);      // global_addr[31:0]
        g0[3] = (unsigned)((ga >> 32) & 0x01FFFFFFull) | (2u << 30); // [56:32]|type=2

        u32x8 g1;                                    // D# group 1 (ISA 8.4)
        g1[0] = 0x00020000u;                         // wg_mask=0, data_size=4B
        g1[1] = ((unsigned)KTOT & 0xFFFFu) << 16;    // tensor_dim0[15:0]
        g1[2] = 1u << 16;                            // tensor_dim0 hi=0, tensor_dim1=1
        g1[3] = ((unsigned)KTOT & 0xFFFFu) << 16;    // tensor_dim1 hi=0, tile_dim0=KTOT
        g1[4] = 1u;                                  // tile_dim1=1, tile_dim2=0
        g1[5] = (unsigned)KTOT;                      // tensor_dim0_stride[31:0]
        g1[6] = 0u;                                  // stride hi / dim1_stride lo
        g1[7] = 0u;

        asm volatile("tensor_load_to_lds %0, %1" :: "s"(g0), "s"(g1) : "memory");
        __builtin_amdgcn_s_wait_tensorcnt(0);        // TENSORcnt==0: tile in LDS
    }

    for (int j = tid; j < KTOT; j += bdim) {
        int lvl = (j >= 1000) + (j >= 2000) + (j >= 3000) + (j >= 4000);
        float ov = (float)lvl * (maxc + 1.0f);
        s_off[j] = ov;
        float4 q = bb[j];
        // reference computes areas on the offset boxes: ((x2+o)-(x1+o))*((y2+o)-(y1+o))
        s_area[j] = ((q.z + ov) - (q.x + ov)) * ((q.w + ov) - (q.y + ov));
    }
    __syncthreads();

    int* __restrict__ kout = keep + img * POSTN;
    const int wave = tid >> 5, lane = tid & 31;

    for (int it = 0; it < POSTN; ++it) {
        // ---- block argmax over s_sc ----
        float bv = -INFINITY; int bi = 0;
        for (int j = tid; j < KTOT; j += bdim) {
            float s = s_sc[j];
            if (s > bv) { bv = s; bi = j; }
        }
        for (int o = 16; o > 0; o >>= 1) {           // wave32 shuffle reduce
            float ov = __shfl_down(bv, o, 32);
            int   oi = __shfl_down(bi, o, 32);
            if (ov > bv) { bv = ov; bi = oi; }
        }
        if (lane == 0) { s_redV[wave] = bv; s_redI[wave] = bi; }
        __syncthreads();
        if (wave == 0) {
            bv = s_redV[lane]; bi = s_redI[lane];
            for (int o = 16; o > 0; o >>= 1) {
                float ov = __shfl_down(bv, o, 32);
                int   oi = __shfl_down(bi, o, 32);
                if (ov > bv) { bv = ov; bi = oi; }
            }
            if (lane == 0) { s_bestI = bi; s_bestV = bv; }
        }
        __syncthreads();

        if (!(s_bestV > -INFINITY)) {                // exhausted: pad with -1
            for (int t2 = it + tid; t2 < POSTN; t2 += bdim) kout[t2] = -1;
            break;                                   // uniform across block
        }

        if (tid == 0) {
            int b = s_bestI;
            kout[it] = b;
            float4 q = bb[b];
            float ob = s_off[b];
            s_b[0] = q.x + ob; s_b[1] = q.y + ob;
            s_b[2] = q.z + ob; s_b[3] = q.w + ob;
            s_b[4] = s_area[b];
            s_sc[b] = -INFINITY;                     // self-suppress (ref: .at[j].set)
        }
        __syncthreads();

        const float bx1 = s_b[0], by1 = s_b[1], bx2 = s_b[2], by2 = s_b[3], ba = s_b[4];
        for (int j = tid; j < KTOT; j += bdim) {
            if (s_sc[j] == -INFINITY) continue;      // suppressed stays suppressed
            float4 q = bb[j];
            float oj = s_off[j];
            float ix1 = fmaxf(bx1, q.x + oj);
            float iy1 = fmaxf(by1, q.y + oj);
            float ix2 = fminf(bx2, q.z + oj);
            float iy2 = fminf(by2, q.w + oj);
            float inter = fmaxf(ix2 - ix1, 0.0f) * fmaxf(iy2 - iy1, 0.0f);
            float iou = inter / (ba + s_area[j] - inter);
            if (iou > NMS_THR) s_sc[j] = -INFINITY;
        }
        __syncthreads();
    }
}

// ---------------------------------------------------------------------------
// Kernel 4: emit (boxes[16,1000,4], scores[16,1000], vk[16,1000]) flat.
// ---------------------------------------------------------------------------
__global__ void k_out(const float* __restrict__ boxesG,
                      const float* __restrict__ sigS,
                      const int*   __restrict__ keep,
                      float* __restrict__ out) {
    int t = blockIdx.x * blockDim.x + threadIdx.x;
    if (t >= NIMG * POSTN) return;
    const int img = t / POSTN;
    const int kk  = keep[t];
    const bool vk = kk >= 0;
    float4 b; b.x = 0.f; b.y = 0.f; b.z = 0.f; b.w = 0.f;
    float sc = 0.0f;
    if (vk) {
        b  = ((const float4*)boxesG)[img * KTOT + kk];
        sc = sigS[img * KTOT + kk];
    }
    float* ob = out;                       // boxes  [16,1000,4]
    float* os = out + NIMG * POSTN * 4;    // scores [16,1000]
    float* ov = out + NIMG * POSTN * 5;    // vk     [16,1000]
    ((float4*)ob)[t] = b;
    os[t] = sc;
    ov[t] = vk ? 1.0f : 0.0f;
}

// ---------------------------------------------------------------------------
extern "C" void kernel_launch(void* const* d_in, const int* in_sizes, int n_in,
                              void* d_out, int out_size, void* d_ws, size_t ws_size,
                              hipStream_t stream) {
    const float* cls   = (const float*)d_in[0];   // [16, 159882]
    const float* reg   = (const float*)d_in[1];   // [16, 159882, 4]
    const float* anch  = (const float*)d_in[2];   // [159882, 4]
    const float* imgsz = (const float*)d_in[3];   // [16, 2]
    float* out = (float*)d_out;

    // Workspace carve (all offsets 16B-aligned: 16*4507*4 = 288448 % 16 == 0)
    char* p = (char*)d_ws;
    int*      sel    = (int*)p;      p += sizeof(int)   * NIMG * KTOT;
    float*    boxesG = (float*)p;    p += sizeof(float) * NIMG * KTOT * 4;
    float*    sigS   = (float*)p;    p += sizeof(float) * NIMG * KTOT;
    float*    smS    = (float*)p;    p += sizeof(float) * NIMG * KTOT;
    unsigned* maxcU  = (unsigned*)p; p += sizeof(unsigned) * 16;  // keep 16B align
    int*      keep   = (int*)p;      p += sizeof(int)   * NIMG * POSTN;
    (void)ws_size; (void)in_sizes; (void)n_in; (void)out_size;

    k_init_max<<<1, 32, 0, stream>>>(maxcU);
    k_topk<<<NIMG * 5, 1024, 0, stream>>>(cls, sel);
    k_decode<<<(NIMG * KTOT + 255) / 256, 256, 0, stream>>>(
        cls, reg, anch, imgsz, sel, boxesG, sigS, smS, maxcU);
    k_nms<<<NIMG, 1024, 0, stream>>>(boxesG, smS, maxcU, keep);
    k_out<<<(NIMG * POSTN + 255) / 256, 256, 0, stream>>>(boxesG, sigS, keep, out);
}